// SSIMLoss_80788334838292
// MI455X (gfx1250) — compile-verified
//
#include <hip/hip_runtime.h>

// SSIM loss, (16,3,512,512) fp32. Depthwise (1,11) Gaussian conv, pad=5 on H and W.
// H-pad output rows are analytically ssim==1 (conv of zero rows) -> added in finalize.
// Conv-as-matmul on V_WMMA_F32_16X16X4_F32: 16x16 tile = A(16x32) x B(32x16),
// B = banded Gaussian. LDS-staged inputs: coalesced b128 global loads (async-to-LDS
// when the builtin exists), conflict-free LDS reads (stride 164 dwords).
// Deterministic two-stage reduction (no atomics).

typedef __attribute__((ext_vector_type(2))) float v2f;
typedef __attribute__((ext_vector_type(8))) float v8f;
typedef __attribute__((ext_vector_type(4))) int   v4i;

typedef __attribute__((address_space(1))) v4i gv4i;   // global int4
typedef __attribute__((address_space(3))) v4i lv4i;   // LDS int4

#define HH          512
#define WW          512
#define NBC         48                      // 16*3 images
#define NBLOCKS     (NBC * 32 * 4)          // 6144: (bc, rowTile, colGroup)
#define LDS_STRIDE  164                     // dwords; 164%64=36 -> conflict-free
#define LDS_COLS    160                     // staged cols: [x0-16, x0+144)
#define C1F         1.0e-4f                 // 0.01^2
#define C2F         9.0e-4f                 // 0.03^2
#define TOTAL_ELEMS 12828672.0              // 48 * (512+10) * 512
#define PAD_ELEMS   245760.0                // 48 * 10 * 512, ssim==1 each

#ifdef __has_builtin
#if __has_builtin(__builtin_amdgcn_global_load_async_to_lds_b128) && \
    __has_builtin(__builtin_amdgcn_s_wait_asynccnt)
#define USE_ASYNC_LDS 1
#endif
#endif
#ifndef USE_ASYNC_LDS
#define USE_ASYNC_LDS 0
#endif

// Normalized 11-tap Gaussian, sigma=1.5 (matches reference _gaussian_window)
__device__ __constant__ float G11[11] = {
    0.00102838f, 0.00759874f, 0.03600077f, 0.10936069f, 0.21300553f,
    0.26601172f,
    0.21300553f, 0.10936069f, 0.03600077f, 0.00759874f, 0.00102838f
};

__device__ __forceinline__ float band(int d) {
    return (d >= 0 && d <= 10) ? G11[d] : 0.0f;
}

__global__ __launch_bounds__(256)
void ssim_main(const float* __restrict__ pred,
               const float* __restrict__ targ,
               double* __restrict__ partial) {
    __shared__ __align__(16) float lds[2 * 16 * LDS_STRIDE]; // pred band, targ band
    __shared__ float wsum[8];

    const int tid  = threadIdx.x;
    const int lane = tid & 31;
    const int w    = tid >> 5;              // wave in block: col-tile 0..7

    const int bid  = blockIdx.x;
    const int colG = bid & 3;               // 128-col group
    const int rowT = (bid >> 2) & 31;
    const int bc   = bid >> 7;
    const int x0   = colG << 7;
    const int row0 = rowT << 4;

    // ---- Stage 16 rows x 160 cols of pred and targ into LDS (zero W-OOB) ----
    // 2 tensors * 16 rows * 40 quads = 1280 float4 = exactly 5 per thread.
    const size_t imgBase = (size_t)bc * HH * WW;
#pragma unroll
    for (int it = 0; it < 5; ++it) {
        const int idx  = tid + it * 256;        // 0..1279
        const int tsel = (idx >= 640) ? 1 : 0;
        const int j    = idx - (tsel ? 640 : 0);
        const int r    = j / 40;
        const int q    = j - r * 40;
        const int x    = x0 - 16 + (q << 2);    // quad 4-aligned: fully in or out
        float* dst = lds + tsel * (16 * LDS_STRIDE) + r * LDS_STRIDE + (q << 2);
        const float* src = (tsel ? targ : pred) + imgBase + (size_t)(row0 + r) * WW + x;
        if ((unsigned)x < (unsigned)WW) {
#if USE_ASYNC_LDS
            __builtin_amdgcn_global_load_async_to_lds_b128(
                (gv4i*)src, (lv4i*)dst, 0, 0);
#else
            const float4 v = *(const float4*)src;
            *(float4*)dst = v;
#endif
        } else {
            dst[0] = 0.f; dst[1] = 0.f; dst[2] = 0.f; dst[3] = 0.f;
        }
    }
#if USE_ASYNC_LDS
    __builtin_amdgcn_s_wait_asynccnt(0);
#endif
    __syncthreads();

    // ---- B fragments: banded Gaussian, Wb[k][n] = g[k-n] for 0<=k-n<=10 ----
    // f32 4x16 B layout: lane = N, lane-half selects K-pair, vgpr idx within pair.
    const int n     = lane & 15;
    const int khalf = (lane >> 4) << 1;
    v2f bfrag[8];
#pragma unroll
    for (int s = 0; s < 8; ++s) {
        const int k = 4 * s + khalf;
        bfrag[s].x = band(k     - n);
        bfrag[s].y = band(k + 1 - n);
    }

    // ---- 8 chained K=4 WMMA steps x 5 quantities; A read from LDS ----
    // A layout (f32 16x4): lane 0-15 = row M, K = 4s + khalf + {0,1}.
    // Global x for K=k is colBase-5+k; LDS col = x-(x0-16) = 16w+11+4s+khalf.
    const float* ldsP = lds + (lane & 15) * LDS_STRIDE;
    const float* ldsT = ldsP + 16 * LDS_STRIDE;
    const int cw = (w << 4) + 11;

    v8f accP = {}, accT = {}, accPP = {}, accTT = {}, accPT = {};
#pragma unroll
    for (int s = 0; s < 8; ++s) {
        const int c = cw + 4 * s + khalf;
        const float p0 = ldsP[c], p1 = ldsP[c + 1];
        const float t0 = ldsT[c], t1 = ldsT[c + 1];

        const v2f aP  = { p0,      p1      };
        const v2f aT  = { t0,      t1      };
        const v2f aPP = { p0 * p0, p1 * p1 };
        const v2f aTT = { t0 * t0, t1 * t1 };
        const v2f aPT = { p0 * t0, p1 * t1 };
        const v2f b   = bfrag[s];

        accP  = __builtin_amdgcn_wmma_f32_16x16x4_f32(false, aP,  false, b, (short)0, accP,  false, false);
        accT  = __builtin_amdgcn_wmma_f32_16x16x4_f32(false, aT,  false, b, (short)0, accT,  false, false);
        accPP = __builtin_amdgcn_wmma_f32_16x16x4_f32(false, aPP, false, b, (short)0, accPP, false, false);
        accTT = __builtin_amdgcn_wmma_f32_16x16x4_f32(false, aTT, false, b, (short)0, accTT, false, false);
        accPT = __builtin_amdgcn_wmma_f32_16x16x4_f32(false, aPT, false, b, (short)0, accPT, false, false);
    }

    // ---- Per-element SSIM; layout-agnostic since we only sum the tile ----
    float tsum = 0.f;
#pragma unroll
    for (int j = 0; j < 8; ++j) {
        const float mu1  = accP[j], mu2 = accT[j];
        const float mu1s = mu1 * mu1;
        const float mu2s = mu2 * mu2;
        const float mu12 = mu1 * mu2;
        const float s1   = accPP[j] - mu1s;
        const float s2   = accTT[j] - mu2s;
        const float s12  = accPT[j] - mu12;
        const float num  = (2.f * mu12 + C1F) * (2.f * s12 + C2F);
        const float den  = (mu1s + mu2s + C1F) * (s1 + s2 + C2F);
        tsum += num / den;
    }
#pragma unroll
    for (int off = 16; off > 0; off >>= 1)
        tsum += __shfl_xor(tsum, off, 32);

    if (lane == 0) wsum[w] = tsum;
    __syncthreads();
    if (tid == 0) {
        double bs = 0.0;
#pragma unroll
        for (int i = 0; i < 8; ++i) bs += (double)wsum[i];
        partial[bid] = bs;                  // overwritten every call: deterministic
    }
}

__global__ __launch_bounds__(256)
void ssim_finalize(const double* __restrict__ partial, float* __restrict__ out) {
    __shared__ double sred[8];
    double a = 0.0;
    for (int i = threadIdx.x; i < NBLOCKS; i += 256) a += partial[i];
#pragma unroll
    for (int off = 16; off > 0; off >>= 1)
        a += __shfl_xor(a, off, 32);
    if ((threadIdx.x & 31) == 0) sred[threadIdx.x >> 5] = a;
    __syncthreads();
    if (threadIdx.x == 0) {
        double t = 0.0;
#pragma unroll
        for (int i = 0; i < 8; ++i) t += sred[i];
        out[0] = (float)(1.0 - (t + PAD_ELEMS) / TOTAL_ELEMS);
    }
}

extern "C" void kernel_launch(void* const* d_in, const int* in_sizes, int n_in,
                              void* d_out, int out_size, void* d_ws, size_t ws_size,
                              hipStream_t stream) {
    const float* pred = (const float*)d_in[0];
    const float* targ = (const float*)d_in[1];
    float*  out     = (float*)d_out;
    double* partial = (double*)d_ws;        // 6144 doubles = 48 KB scratch

    ssim_main<<<NBLOCKS, 256, 0, stream>>>(pred, targ, partial);
    ssim_finalize<<<1, 256, 0, stream>>>(partial, out);
}